// GraphDecoder_87333864997320
// MI455X (gfx1250) — compile-verified
//
#include <hip/hip_runtime.h>
#include <math.h>

typedef __attribute__((ext_vector_type(2))) float v2f;
typedef __attribute__((ext_vector_type(8))) float v8f;

#define LATENT 64
#define HIDDEN 96
#define OUTDIM 64

// ---------------- degree / normalization ----------------

__global__ void k_deg_init(float* deg, int n) {
  int i = blockIdx.x * blockDim.x + threadIdx.x;
  if (i < n) deg[i] = 1.0f;  // self-loop
}

__global__ void k_deg_accum(const int* __restrict__ dst, float* deg, int E) {
  int e = blockIdx.x * blockDim.x + threadIdx.x;
  if (e < E) atomicAdd(&deg[dst[e]], 1.0f);
}

__global__ void k_deg_to_dis(float* deg, int n) {
  int i = blockIdx.x * blockDim.x + threadIdx.x;
  if (i < n) deg[i] = rsqrtf(deg[i]);
}

// ---------------- fp32 WMMA GEMM: C[M,N] = A[M,K] x B[K,N] ----------------
// Requires M%16==0, N%16==0, K%4==0 (true for all calls here).
// One wave computes one 16x16 output tile via V_WMMA_F32_16X16X4_F32.
__global__ void k_gemm_wmma_f32(const float* __restrict__ A,
                                const float* __restrict__ B,
                                float* __restrict__ C,
                                int M, int N, int K) {
  int wave = (blockIdx.x * blockDim.x + threadIdx.x) >> 5;
  int ntn = N >> 4;
  int ntiles = (M >> 4) * ntn;
  if (wave >= ntiles) return;  // wave-uniform: EXEC all-1s inside

  int trow = wave / ntn;
  int tcol = wave - trow * ntn;
  int row0 = trow << 4;
  int col0 = tcol << 4;

  int lane = threadIdx.x & 31;
  int half = lane >> 4;   // 0: K pair {0,1}, 1: K pair {2,3} within each 4-step
  int lr   = lane & 15;   // M index for A frag, N index for B frag

  v8f acc = {0.f, 0.f, 0.f, 0.f, 0.f, 0.f, 0.f, 0.f};
  const float* arow = A + (size_t)(row0 + lr) * K;

  for (int k = 0; k < K; k += 4) {
    v2f a, b;
    a.x = arow[k + 2 * half + 0];
    a.y = arow[k + 2 * half + 1];
    b.x = B[(size_t)(k + 2 * half + 0) * N + col0 + lr];
    b.y = B[(size_t)(k + 2 * half + 1) * N + col0 + lr];
    acc = __builtin_amdgcn_wmma_f32_16x16x4_f32(
        /*neg_a=*/false, a, /*neg_b=*/false, b,
        /*c_mod=*/(short)0, acc, /*reuse_a=*/false, /*reuse_b=*/false);
  }

#pragma unroll
  for (int r = 0; r < 8; ++r)
    C[(size_t)(row0 + r + 8 * half) * N + col0 + lr] = acc[r];
}

// ---------------- aggregation ----------------

// agg[i,j] = h[i,j] * dis[i]^2   (self-loop term, also initializes agg)
__global__ void k_self_init(const float* __restrict__ h,
                            const float* __restrict__ dis,
                            float* __restrict__ agg, int n, int F) {
  int gid = blockIdx.x * blockDim.x + threadIdx.x;
  if (gid >= n * F) return;
  int i = gid / F;
  float d = dis[i];
  agg[gid] = h[gid] * d * d;
}

// per edge: agg[dst] += h[src] * dis[src]*dis[dst]   (float4-vectorized feats)
__global__ void k_scatter(const float* __restrict__ h,
                          const float* __restrict__ dis,
                          const int* __restrict__ src,
                          const int* __restrict__ dst,
                          float* agg, int E, int F) {
  int fq = F >> 2;
  int gid = blockIdx.x * blockDim.x + threadIdx.x;
  if (gid >= E * fq) return;
  int e = gid / fq;
  int j = (gid - e * fq) << 2;
  int s = src[e];
  int d = dst[e];
  float nrm = dis[s] * dis[d];
  const float4* hp = (const float4*)(h + (size_t)s * F + j);
  float4 v = *hp;
  float* ap = agg + (size_t)d * F + j;
  atomicAdd(ap + 0, v.x * nrm);
  atomicAdd(ap + 1, v.y * nrm);
  atomicAdd(ap + 2, v.z * nrm);
  atomicAdd(ap + 3, v.w * nrm);
}

__global__ void k_bias_relu(const float* __restrict__ a,
                            const float* __restrict__ bias,
                            float* __restrict__ o, int tot, int F) {
  int gid = blockIdx.x * blockDim.x + threadIdx.x;
  if (gid >= tot) return;
  int j = gid % F;
  float v = a[gid] + bias[j];
  o[gid] = v > 0.f ? v : 0.f;
}

__global__ void k_bias_sigmoid(const float* __restrict__ a,
                               const float* __restrict__ bias,
                               float* __restrict__ o, int tot, int F) {
  int gid = blockIdx.x * blockDim.x + threadIdx.x;
  if (gid >= tot) return;
  int j = gid % F;
  float v = a[gid] + bias[j];
  o[gid] = 1.0f / (1.0f + __expf(-v));
}

// ---------------- launch ----------------

extern "C" void kernel_launch(void* const* d_in, const int* in_sizes, int n_in,
                              void* d_out, int out_size, void* d_ws, size_t ws_size,
                              hipStream_t stream) {
  const float* x  = (const float*)d_in[0];
  const int*   ei = (const int*)d_in[1];
  const float* W1 = (const float*)d_in[2];
  const float* b1 = (const float*)d_in[3];
  const float* W2 = (const float*)d_in[4];
  const float* b2 = (const float*)d_in[5];
  float* out = (float*)d_out;

  int n = in_sizes[0] / LATENT;   // 50000
  int E = in_sizes[1] / 2;        // 800000
  const int* src = ei;
  const int* dst = ei + E;

  // workspace layout (floats): dis[n] | hraw[n*96] | agg[n*96] | h2raw[n*64]
  float* dis   = (float*)d_ws;
  float* hraw  = dis + n;
  float* agg   = hraw + (size_t)n * HIDDEN;
  float* h2raw = agg + (size_t)n * HIDDEN;

  const int B = 256;
  // --- normalization coefficients ---
  k_deg_init<<<(n + B - 1) / B, B, 0, stream>>>(dis, n);
  k_deg_accum<<<(E + B - 1) / B, B, 0, stream>>>(dst, dis, E);
  k_deg_to_dis<<<(n + B - 1) / B, B, 0, stream>>>(dis, n);

  // --- layer 1: h = relu(norm_agg(x @ W1) + b1) ---
  {
    int tiles = (n / 16) * (HIDDEN / 16);
    int thr = tiles * 32;
    k_gemm_wmma_f32<<<(thr + B - 1) / B, B, 0, stream>>>(x, W1, hraw, n, HIDDEN, LATENT);
    int tot = n * HIDDEN;
    k_self_init<<<(tot + B - 1) / B, B, 0, stream>>>(hraw, dis, agg, n, HIDDEN);
    int eg = E * (HIDDEN / 4);
    k_scatter<<<(eg + B - 1) / B, B, 0, stream>>>(hraw, dis, src, dst, agg, E, HIDDEN);
    k_bias_relu<<<(tot + B - 1) / B, B, 0, stream>>>(agg, b1, hraw, tot, HIDDEN);
  }

  // --- layer 2: out = sigmoid(norm_agg(h @ W2) + b2), aggregated in d_out ---
  {
    int tiles = (n / 16) * (OUTDIM / 16);
    int thr = tiles * 32;
    k_gemm_wmma_f32<<<(thr + B - 1) / B, B, 0, stream>>>(hraw, W2, h2raw, n, OUTDIM, HIDDEN);
    int tot = n * OUTDIM;
    k_self_init<<<(tot + B - 1) / B, B, 0, stream>>>(h2raw, dis, out, n, OUTDIM);
    int eg = E * (OUTDIM / 4);
    k_scatter<<<(eg + B - 1) / B, B, 0, stream>>>(h2raw, dis, src, dst, out, E, OUTDIM);
    k_bias_sigmoid<<<(tot + B - 1) / B, B, 0, stream>>>(out, b2, out, tot, OUTDIM);
  }
}